// ConstrainLoss_57475252355294
// MI455X (gfx1250) — compile-verified
//
#include <hip/hip_runtime.h>
#include <math.h>

// ---------------------------------------------------------------------------
// Fused:  loss = -dot( logsumexp_j( -max(||x_i-o_j||^2,0)*inv2s2 + log(w_j) ), x_w )
// N = M = 16384, d = 32, inv2s2 = 1/(2*(0.5)^2) = 2.0
//
// Pass 1: per-column (||o_j||^2, log w_j) pairs -> d_ws   (128 KB, L2-resident)
// Pass 2: one 16-row tile per block; 4 waves/block split the 1024 column tiles.
//         GEMM via V_WMMA_F32_16X16X4_F32, online logsumexp fused in registers.
// Pass 3: deterministic tree reduction of the 1024 block partials.
// ---------------------------------------------------------------------------

typedef __attribute__((ext_vector_type(2))) float v2f;
typedef __attribute__((ext_vector_type(4))) float v4f;
typedef __attribute__((ext_vector_type(8))) float v8f;

#define DIMK 32
#define TILE 16
#define WAVES_PER_BLOCK 4
#define INV_TWO_S2 2.0f

// ---- pass 1: column norms + log-weights, computed once ---------------------
__global__ __launch_bounds__(256) void cl_colprep_kernel(
    const float* __restrict__ xo,    // [M, 32]
    const float* __restrict__ xo_w,  // [M]
    v2f* __restrict__ onlw,          // [M] pairs (||o_j||^2, log w_j)
    int M) {
  const int j = blockIdx.x * blockDim.x + threadIdx.x;
  if (j >= M) return;
  const v4f* r = (const v4f*)(xo + (size_t)j * DIMK);
  float s = 0.f;
#pragma unroll
  for (int c = 0; c < 8; ++c) {
    const v4f v = r[c];
    s = fmaf(v.x, v.x, s);
    s = fmaf(v.y, v.y, s);
    s = fmaf(v.z, v.z, s);
    s = fmaf(v.w, v.w, s);
  }
  v2f o;
  o.x = s;
  o.y = __logf(xo_w[j]);
  onlw[j] = o;
}

// ---- pass 2: fused WMMA distance + online logsumexp ------------------------
__global__ __launch_bounds__(32 * WAVES_PER_BLOCK) void cl_main_kernel(
    const float* __restrict__ x,      // [N, 32]
    const float* __restrict__ x_w,    // [N]
    const float* __restrict__ xo,     // [M, 32]
    const v2f* __restrict__ onlw,     // [M] (||o_j||^2, log w_j)
    float* __restrict__ partial,      // [N/16] per-block sums of x_w[i]*lse_i
    int M) {
  __shared__ float lm[WAVES_PER_BLOCK][16];
  __shared__ float ls[WAVES_PER_BLOCK][16];

  const int lane = threadIdx.x & 31;   // wave32
  const int wave = threadIdx.x >> 5;
  const int half = lane >> 4;          // 0: K%4 in {0,1}, rows 0-7 of D ; 1: K%4 in {2,3}, rows 8-15
  const int c16  = lane & 15;
  const int rowBase = blockIdx.x * TILE;

  // A tile (16 rows x 32) once, in f32 16x16x4 WMMA layout (8 K-chunks)
  const v2f* arow = (const v2f*)(x + (size_t)(rowBase + c16) * DIMK);
  v2f a[8];
#pragma unroll
  for (int c = 0; c < 8; ++c) a[c] = arow[2 * c + half];

  // row norms ||x_i||^2 (once per block; bpermute here is off the hot path)
  float pa = 0.f;
#pragma unroll
  for (int c = 0; c < 8; ++c) {
    pa = fmaf(a[c].x, a[c].x, pa);
    pa = fmaf(a[c].y, a[c].y, pa);
  }
  const float fulln = pa + __shfl_xor(pa, 16, 32);  // lane L: norm of row rowBase+(L&15)
  float xn[8];
#pragma unroll
  for (int v = 0; v < 8; ++v) xn[v] = __shfl(fulln, half * 8 + v, 32);  // row = 8*half+v (D layout)

  // online logsumexp state: 8 rows per lane, column == c16 within each tile
  float m[8], s[8];
#pragma unroll
  for (int v = 0; v < 8; ++v) { m[v] = -__builtin_inff(); s[v] = 0.f; }

  const int tilesPerWave = (M / TILE) / WAVES_PER_BLOCK;
  const int ctBase = wave * tilesPerWave;

  for (int t = 0; t < tilesPerWave; ++t) {
    const int col = (ctBase + t) * TILE + c16;

    // prefetch next column tile (gfx1250 global_prefetch path)
    if (t + 1 < tilesPerWave)
      __builtin_prefetch(xo + (size_t)(col + TILE) * DIMK, 0, 0);

    // B tile: x_obs rows as columns of B (4x16, K-major striping)
    const v2f* brow = (const v2f*)(xo + (size_t)col * DIMK);
    v2f b[8];
#pragma unroll
    for (int c = 0; c < 8; ++c) b[c] = brow[2 * c + half];

    // per-column constants: one 64-bit load replaces 16 FMAs + bpermute + log
    const v2f ol = onlw[col];
    const float on = ol.x;
    const float lw = ol.y;

    // dot(x_i, o_j) via 8x V_WMMA_F32_16X16X4_F32
    v8f acc = {0.f, 0.f, 0.f, 0.f, 0.f, 0.f, 0.f, 0.f};
#pragma unroll
    for (int c = 0; c < 8; ++c)
      acc = __builtin_amdgcn_wmma_f32_16x16x4_f32(
          false, a[c], false, b[c], (short)0, acc, false, false);

    // xn[v] + on once per tile (packs well), then per-element epilogue:
    //   val = min(2*dot - (xn+on), 0) * 2 + lw   (fma-folded)
    float xnon[8];
#pragma unroll
    for (int v = 0; v < 8; ++v) xnon[v] = xn[v] + on;

#pragma unroll
    for (int v = 0; v < 8; ++v) {
      const float q   = fminf(fmaf(2.f, acc[v], -xnon[v]), 0.f);
      const float val = fmaf(q, INV_TWO_S2, lw);
      const float nm  = fmaxf(m[v], val);
      const float e   = __expf(-fabsf(val - m[v]));  // single v_exp_f32 / element
      const bool  gt  = val > m[v];
      s[v] = gt ? fmaf(s[v], e, 1.f) : (s[v] + e);
      m[v] = nm;
    }
  }

  // merge LSE state across the 16 lanes holding the same row
#pragma unroll
  for (int v = 0; v < 8; ++v) {
#pragma unroll
    for (int mask = 8; mask >= 1; mask >>= 1) {
      const float om = __shfl_xor(m[v], mask, 32);
      const float os = __shfl_xor(s[v], mask, 32);
      const float nm = fmaxf(m[v], om);
      s[v] = s[v] * __expf(m[v] - nm) + os * __expf(om - nm);
      m[v] = nm;
    }
  }

  // lanes 0 and 16 hold the reduced (m,s) for rows 8*half+0..7
  if (c16 == 0) {
#pragma unroll
    for (int v = 0; v < 8; ++v) {
      lm[wave][half * 8 + v] = m[v];
      ls[wave][half * 8 + v] = s[v];
    }
  }
  __syncthreads();

  // merge across the 4 waves (disjoint column ranges), finish rows
  if (threadIdx.x < 16) {
    const int r = threadIdx.x;
    float M0 = lm[0][r], S0 = ls[0][r];
#pragma unroll
    for (int w = 1; w < WAVES_PER_BLOCK; ++w) {
      const float mm = fmaxf(M0, lm[w][r]);
      S0 = S0 * __expf(M0 - mm) + ls[w][r] * __expf(lm[w][r] - mm);
      M0 = mm;
    }
    float contrib = x_w[rowBase + r] * (M0 + __logf(S0));
#pragma unroll
    for (int mask = 8; mask >= 1; mask >>= 1)
      contrib += __shfl_xor(contrib, mask, 32);
    if (r == 0) partial[blockIdx.x] = contrib;
  }
}

// ---- pass 3: deterministic final reduction ---------------------------------
__global__ __launch_bounds__(256) void cl_reduce_kernel(
    const float* __restrict__ partial, float* __restrict__ out, int n) {
  __shared__ float sm[256];
  float acc = 0.f;
  for (int i = threadIdx.x; i < n; i += 256) acc += partial[i];
  sm[threadIdx.x] = acc;
  __syncthreads();
#pragma unroll
  for (int off = 128; off > 0; off >>= 1) {
    if ((int)threadIdx.x < off) sm[threadIdx.x] += sm[threadIdx.x + off];
    __syncthreads();
  }
  if (threadIdx.x == 0) out[0] = -sm[0];
}

extern "C" void kernel_launch(void* const* d_in, const int* in_sizes, int n_in,
                              void* d_out, int out_size, void* d_ws, size_t ws_size,
                              hipStream_t stream) {
  const float* x    = (const float*)d_in[0];  // [N,32]
  const float* x_w  = (const float*)d_in[1];  // [N]
  const float* xo   = (const float*)d_in[2];  // [M,32]
  const float* xo_w = (const float*)d_in[3];  // [M]
  float* out = (float*)d_out;

  const int N = in_sizes[1];
  const int M = in_sizes[3];
  const int nBlocks = N / TILE;  // 1024

  // scratch layout: [2*M floats (on,lw) pairs][nBlocks floats partials]
  v2f*   onlw    = (v2f*)d_ws;
  float* partial = (float*)d_ws + (size_t)2 * M;

  cl_colprep_kernel<<<(M + 255) / 256, 256, 0, stream>>>(xo, xo_w, onlw, M);
  cl_main_kernel<<<nBlocks, 32 * WAVES_PER_BLOCK, 0, stream>>>(
      x, x_w, xo, onlw, partial, M);
  cl_reduce_kernel<<<1, 256, 0, stream>>>(partial, out, nBlocks);
}